// Prompt_12094627905989
// MI455X (gfx1250) — compile-verified
//
#include <hip/hip_runtime.h>
#include <hip/hip_bf16.h>

// ---------------- problem constants ----------------
constexpr int B_    = 256;
constexpr int L_    = 512;
constexpr int D_    = 768;
constexpr int P_    = 30;   // pool size
constexpr int PPAD_ = 32;   // padded pool for WMMA N tiles
constexpr int K_    = 4;    // top-k
constexpr int LOUT_ = L_ + K_;   // 516

constexpr int    V4_    = D_ / 4;                      // 192 float4 per row
constexpr size_t EMB_N  = (size_t)B_ * LOUT_ * D_;     // prompted_embedding elems
constexpr size_t SIM_OFF = EMB_N;                      // similarity [B,30]
constexpr size_t RS_OFF  = SIM_OFF + (size_t)B_ * P_;  // reduce_sim scalar
constexpr size_t IDX_OFF = RS_OFF + 1;                 // idx [B,4] (int32 bits)

typedef float v2f __attribute__((ext_vector_type(2)));
typedef float v8f __attribute__((ext_vector_type(8)));

// ---------------------------------------------------------------------------
// Kernel 1: per-batch fused  (mean over L) + (copy x_embed into output tail)
//           + L2-normalize mean -> x_norm (workspace).
// One block per batch, 192 threads, float4 lanes. Single pass over 402 MB.
// ---------------------------------------------------------------------------
__global__ void mean_copy_norm_kernel(const float* __restrict__ x,
                                      float* __restrict__ out,
                                      float* __restrict__ xn) {
    const int b = blockIdx.x;
    const int t = threadIdx.x;                       // 0..191
    const float4* src = (const float4*)(x + (size_t)b * L_ * D_);
    float4*       dst = (float4*)(out + (size_t)b * LOUT_ * D_ + (size_t)K_ * D_);

    float ax = 0.f, ay = 0.f, az = 0.f, aw = 0.f;
    #pragma unroll 4
    for (int l = 0; l < L_; ++l) {
        float4 v = src[l * V4_ + t];
        dst[l * V4_ + t] = v;                        // concatenate tail copy
        ax += v.x; ay += v.y; az += v.z; aw += v.w;
    }
    const float inv = 1.0f / (float)L_;
    ax *= inv; ay *= inv; az *= inv; aw *= inv;

    __shared__ float red[192];
    red[t] = ax * ax + ay * ay + az * az + aw * aw;
    __syncthreads();
    if (t < 96) red[t] += red[t + 96]; __syncthreads();
    if (t < 48) red[t] += red[t + 48]; __syncthreads();
    if (t < 24) red[t] += red[t + 24]; __syncthreads();
    if (t < 12) red[t] += red[t + 12]; __syncthreads();
    if (t <  6) red[t] += red[t +  6]; __syncthreads();
    if (t <  3) red[t] += red[t +  3]; __syncthreads();
    const float ssq = red[0] + red[1] + red[2];
    const float rs  = rsqrtf(fmaxf(ssq, 1e-12f));

    float4 o; o.x = ax * rs; o.y = ay * rs; o.z = az * rs; o.w = aw * rs;
    ((float4*)(xn + (size_t)b * D_))[t] = o;
}

// ---------------------------------------------------------------------------
// Kernel 2: L2-normalize prompt rows into padded [32][768] buffer (rows 30,31
// zeroed) so WMMA B-tiles need no predication.
// ---------------------------------------------------------------------------
__global__ void prompt_norm_kernel(const float* __restrict__ prompt,
                                   float* __restrict__ pn) {
    const int p = blockIdx.x;                        // 0..31
    const int t = threadIdx.x;                       // 0..191
    float4* dst = (float4*)(pn + (size_t)p * D_);
    if (p >= P_) { float4 z = {0.f, 0.f, 0.f, 0.f}; dst[t] = z; return; }

    float4 v = ((const float4*)(prompt + (size_t)p * D_))[t];
    __shared__ float red[192];
    red[t] = v.x * v.x + v.y * v.y + v.z * v.z + v.w * v.w;
    __syncthreads();
    if (t < 96) red[t] += red[t + 96]; __syncthreads();
    if (t < 48) red[t] += red[t + 48]; __syncthreads();
    if (t < 24) red[t] += red[t + 24]; __syncthreads();
    if (t < 12) red[t] += red[t + 12]; __syncthreads();
    if (t <  6) red[t] += red[t +  6]; __syncthreads();
    if (t <  3) red[t] += red[t +  3]; __syncthreads();
    const float ssq = red[0] + red[1] + red[2];
    const float rs  = rsqrtf(fmaxf(ssq, 1e-12f));
    float4 o; o.x = v.x * rs; o.y = v.y * rs; o.z = v.z * rs; o.w = v.w * rs;
    dst[t] = o;
}

// ---------------------------------------------------------------------------
// Kernel 3: similarity[256,30] = Xn[256,768] x Pn^T via V_WMMA_F32_16X16X4_F32.
// One wave32 per block; grid = (16 M-tiles, 2 N-tiles). K loop = 192 WMMAs.
// A layout: lanes 0-15 = rows M, v0:K=0 v1:K=1; lanes 16-31: v0:K=2 v1:K=3.
// B layout mirrors A with N across lanes. C: VGPR j -> row j+8*half, col r.
// ---------------------------------------------------------------------------
__global__ void sim_wmma_kernel(const float* __restrict__ Xn,   // [256][768]
                                const float* __restrict__ Pn,   // [32][768]
                                float* __restrict__ sim) {      // [256][30]
    const int lane = threadIdx.x;        // 0..31
    const int half = lane >> 4;          // 0 or 1
    const int r    = lane & 15;
    const int mb   = blockIdx.x * 16;
    const int nb   = blockIdx.y * 16;

    const float* arow = Xn + (size_t)(mb + r) * D_ + 2 * half;
    const float* brow = Pn + (size_t)(nb + r) * D_ + 2 * half;

    v8f c = {};
    #pragma unroll 8
    for (int k0 = 0; k0 < D_; k0 += 4) {
        v2f a = *(const v2f*)(arow + k0);
        v2f b = *(const v2f*)(brow + k0);
        c = __builtin_amdgcn_wmma_f32_16x16x4_f32(
                /*neg_a=*/false, a, /*neg_b=*/false, b,
                /*c_mod=*/(short)0, c, /*reuse_a=*/false, /*reuse_b=*/false);
    }

    const int col = nb + r;
    if (col < P_) {
        #pragma unroll
        for (int j = 0; j < 8; ++j) {
            sim[(size_t)(mb + j + 8 * half) * P_ + col] = c[j];
        }
    }
}

// ---------------------------------------------------------------------------
// Kernel 4: top-4 selection (jax.lax.top_k semantics: descending, ties ->
// lowest index), gather prompt[idx] into output head, write idx bits, and
// per-batch reduce_sim partial ( = sum of selected similarity values ).
// ---------------------------------------------------------------------------
__global__ void topk_gather_kernel(const float* __restrict__ sim,    // [B][30]
                                   const float* __restrict__ prompt, // [30][768]
                                   float* __restrict__ out,          // base
                                   float* __restrict__ partial) {    // [B]
    const int b = blockIdx.x;
    const int t = threadIdx.x;                       // 0..191

    __shared__ float s_sim[P_];
    __shared__ int   s_idx[K_];
    if (t < P_) s_sim[t] = sim[(size_t)b * P_ + t];
    __syncthreads();

    if (t == 0) {
        unsigned taken = 0u;
        float acc = 0.f;
        for (int k = 0; k < K_; ++k) {
            float bv = -__builtin_inff();
            int   bi = 0;
            for (int j = 0; j < P_; ++j) {
                if (!((taken >> j) & 1u) && s_sim[j] > bv) { bv = s_sim[j]; bi = j; }
            }
            taken |= (1u << bi);
            s_idx[k] = bi;
            acc += bv;
        }
        partial[b] = acc;
        int* idx_out = (int*)(out + IDX_OFF);
        #pragma unroll
        for (int k = 0; k < K_; ++k) idx_out[b * K_ + k] = s_idx[k];
    }
    __syncthreads();

    float4* dst = (float4*)(out + (size_t)b * LOUT_ * D_);
    #pragma unroll
    for (int k = 0; k < K_; ++k) {
        const float4* srcp = (const float4*)(prompt + (size_t)s_idx[k] * D_);
        dst[k * V4_ + t] = srcp[t];
    }
}

// ---------------------------------------------------------------------------
// Kernel 5: deterministic final reduction of 256 partials -> reduce_sim.
// ---------------------------------------------------------------------------
__global__ void reduce_final_kernel(const float* __restrict__ partial,
                                    float* __restrict__ out) {
    __shared__ float red[256];
    const int t = threadIdx.x;
    red[t] = partial[t];
    __syncthreads();
    for (int s = 128; s > 0; s >>= 1) {
        if (t < s) red[t] += red[t + s];
        __syncthreads();
    }
    if (t == 0) out[RS_OFF] = red[0] * (1.0f / (float)B_);
}

// ---------------------------------------------------------------------------
extern "C" void kernel_launch(void* const* d_in, const int* in_sizes, int n_in,
                              void* d_out, int out_size, void* d_ws, size_t ws_size,
                              hipStream_t stream) {
    const float* x      = (const float*)d_in[0];   // [256,512,768]
    const float* prompt = (const float*)d_in[1];   // [30,768]
    float* out = (float*)d_out;

    // workspace layout (floats)
    float* ws      = (float*)d_ws;
    float* xn      = ws;                                  // [256][768]
    float* pn      = xn + (size_t)B_ * D_;                // [32][768]
    float* partial = pn + (size_t)PPAD_ * D_;             // [256]

    mean_copy_norm_kernel<<<B_, 192, 0, stream>>>(x, out, xn);
    prompt_norm_kernel<<<PPAD_, 192, 0, stream>>>(prompt, pn);
    sim_wmma_kernel<<<dim3(B_ / 16, PPAD_ / 16), 32, 0, stream>>>(xn, pn, out + SIM_OFF);
    topk_gather_kernel<<<B_, 192, 0, stream>>>(out + SIM_OFF, prompt, out, partial);
    reduce_final_kernel<<<1, 256, 0, stream>>>(partial, out);
}